// WordAttention_11802570130368
// MI455X (gfx1250) — compile-verified
//
#include <hip/hip_runtime.h>
#include <hip/hip_bf16.h>

typedef float v2f __attribute__((ext_vector_type(2)));
typedef float v8f __attribute__((ext_vector_type(8)));

#define HDIM 128

// ---- order-preserving float<->uint encoding for atomic max ----
__device__ __forceinline__ unsigned f32_order(float f) {
    unsigned u = __float_as_uint(f);
    return (u & 0x80000000u) ? ~u : (u | 0x80000000u);
}
__device__ __forceinline__ float f32_unorder(unsigned u) {
    return __uint_as_float((u & 0x80000000u) ? (u ^ 0x80000000u) : ~u);
}

// =====================================================================
// Pass 0: init scalars (sum = 0, max = encode(-inf))
// =====================================================================
__global__ void init_kernel(float* gsum, unsigned* gmax) {
    *gsum = 0.0f;
    *gmax = 0x007FFFFFu;  // f32_order(-INFINITY)
}

// =====================================================================
// Pass 1: Q/K/V = x @ W^T + b   via V_WMMA_F32_16X16X4_F32
// grid = (N/16, 3); block = 256 (8 waves); wave w owns output cols [16w,16w+16)
// =====================================================================
__global__ __launch_bounds__(256) void qkv_proj_kernel(
    const float* __restrict__ x,
    const float* __restrict__ Wq, const float* __restrict__ bq,
    const float* __restrict__ Wk, const float* __restrict__ bk,
    const float* __restrict__ Wv, const float* __restrict__ bv,
    float* __restrict__ Qout, float* __restrict__ Kout, float* __restrict__ Vout)
{
    __shared__ float xs[16 * 132];  // 16 rows x 128 cols, pad 4 -> conflict-free

    const int n0 = blockIdx.x * 16;
    const int m  = blockIdx.y;  // 0=Q 1=K 2=V

    const float* W; const float* b; float* Out;
    if (m == 0)      { W = Wq; b = bq; Out = Qout; }
    else if (m == 1) { W = Wk; b = bk; Out = Kout; }
    else             { W = Wv; b = bv; Out = Vout; }

    // stage x tile [16 x 128] into LDS (coalesced float4 loads)
    {
        const int r  = threadIdx.x >> 4;         // 0..15
        const int c0 = (threadIdx.x & 15) * 8;   // 0..120
        const float4* src = (const float4*)(x + (size_t)(n0 + r) * HDIM + c0);
        float4 a0 = src[0];
        float4 a1 = src[1];
        float* dst = &xs[r * 132 + c0];
        dst[0] = a0.x; dst[1] = a0.y; dst[2] = a0.z; dst[3] = a0.w;
        dst[4] = a1.x; dst[5] = a1.y; dst[6] = a1.z; dst[7] = a1.w;
    }
    __syncthreads();

    const int lane = threadIdx.x & 31;
    const int wv   = threadIdx.x >> 5;   // wave id 0..7
    const int h0   = wv * 16;            // output column tile
    const int half = lane >> 4;          // 0 or 1
    const int l16  = lane & 15;

    // C/D layout: VGPR j, lanes 0-15 -> (M=j, N=lane); lanes 16-31 -> (M=j+8, N=lane-16)
    // init accumulator with bias (constant along M)
    const float bias = b[h0 + l16];
    v8f c;
    #pragma unroll
    for (int j = 0; j < 8; ++j) c[j] = bias;

    // B[k][h] = W[h][k]; lane = N column, halves hold K k0..k0+1 / k0+2..k0+3
    const float* wrow = W + (size_t)(h0 + l16) * HDIM;

    #pragma unroll 4
    for (int s = 0; s < 32; ++s) {
        const int k0 = s * 4;
        const int kk = k0 + half * 2;
        v2f a, bf;
        // A 16x4 f32: lanes 0-15 M=0..15 K={k0,k0+1}; lanes 16-31 K={k0+2,k0+3}
        a.x = xs[l16 * 132 + kk];
        a.y = xs[l16 * 132 + kk + 1];
        bf.x = wrow[kk];
        bf.y = wrow[kk + 1];
        c = __builtin_amdgcn_wmma_f32_16x16x4_f32(false, a, false, bf,
                                                  (short)0, c, false, false);
    }

    float* outp = Out + (size_t)n0 * HDIM + h0;
    #pragma unroll
    for (int j = 0; j < 8; ++j) {
        outp[(size_t)(j + 8 * half) * HDIM + l16] = c[j];
    }
}

// =====================================================================
// Pass 2: per-edge logit = dot(Q[row],K[col]) / sqrt(H) * w[e]; global max
// one wave per edge
// =====================================================================
__global__ __launch_bounds__(256) void edge_logits_kernel(
    const float* __restrict__ Q, const float* __restrict__ K,
    const int* __restrict__ ei, const float* __restrict__ ew,
    float* __restrict__ logits, unsigned* __restrict__ gmax, int E)
{
    __shared__ unsigned wmax[8];
    const int lane = threadIdx.x & 31;
    const int wid  = threadIdx.x >> 5;
    const int e = blockIdx.x * 8 + wid;

    float logit = -INFINITY;
    if (e < E) {
        const int row = ei[e];
        const int col = ei[E + e];
        const float4 q = ((const float4*)(Q + (size_t)row * HDIM))[lane];
        const float4 k = ((const float4*)(K + (size_t)col * HDIM))[lane];
        float d = q.x * k.x + q.y * k.y + q.z * k.z + q.w * k.w;
        #pragma unroll
        for (int off = 16; off; off >>= 1) d += __shfl_xor(d, off, 32);
        logit = d * 0.08838834764831845f * ew[e];  // 1/sqrt(128)
        if (lane == 0) logits[e] = logit;
    }
    if (lane == 0) wmax[wid] = f32_order(logit);
    __syncthreads();
    if (threadIdx.x == 0) {
        unsigned mx = wmax[0];
        #pragma unroll
        for (int i = 1; i < 8; ++i) mx = max(mx, wmax[i]);
        atomicMax(gmax, mx);
    }
}

// =====================================================================
// Pass 3: logits[i] = exp(logits[i] - max); accumulate global sum
// =====================================================================
__global__ __launch_bounds__(256) void exp_sum_kernel(
    float* __restrict__ logits, float* __restrict__ gsum,
    const unsigned* __restrict__ gmax, int E)
{
    __shared__ float wsum[8];
    const int i = blockIdx.x * blockDim.x + threadIdx.x;
    const float mx = f32_unorder(*gmax);
    float v = 0.0f;
    if (i < E) {
        v = expf(logits[i] - mx);
        logits[i] = v;
    }
    #pragma unroll
    for (int off = 16; off; off >>= 1) v += __shfl_xor(v, off, 32);
    const int lane = threadIdx.x & 31;
    const int wid  = threadIdx.x >> 5;
    if (lane == 0) wsum[wid] = v;
    __syncthreads();
    if (threadIdx.x == 0) {
        float s = 0.0f;
        #pragma unroll
        for (int k = 0; k < 8; ++k) s += wsum[k];
        unsafeAtomicAdd(gsum, s);
    }
}

// =====================================================================
// Pass 4: out[row] += V[col] * (exp / sum); one wave per edge, f32 atomics
// =====================================================================
__global__ __launch_bounds__(256) void scatter_kernel(
    const float* __restrict__ V, const float* __restrict__ logits,
    const float* __restrict__ gsum, const int* __restrict__ ei,
    float* __restrict__ out, int E)
{
    const int lane = threadIdx.x & 31;
    const int wid  = threadIdx.x >> 5;
    const int e = blockIdx.x * 8 + wid;
    if (e >= E) return;

    const float a = logits[e] * (1.0f / *gsum);
    const int row = ei[e];
    const int col = ei[E + e];
    const float4 v4 = ((const float4*)(V + (size_t)col * HDIM))[lane];
    float* dst = out + (size_t)row * HDIM + lane * 4;
    unsafeAtomicAdd(dst + 0, v4.x * a);
    unsafeAtomicAdd(dst + 1, v4.y * a);
    unsafeAtomicAdd(dst + 2, v4.z * a);
    unsafeAtomicAdd(dst + 3, v4.w * a);
}

extern "C" void kernel_launch(void* const* d_in, const int* in_sizes, int n_in,
                              void* d_out, int out_size, void* d_ws, size_t ws_size,
                              hipStream_t stream) {
    const float* x  = (const float*)d_in[0];
    const int*   ei = (const int*)d_in[1];     // [2, E] int32
    const float* ew = (const float*)d_in[2];
    const float* Wq = (const float*)d_in[3];
    const float* bq = (const float*)d_in[4];
    const float* Wk = (const float*)d_in[5];
    const float* bk = (const float*)d_in[6];
    const float* Wv = (const float*)d_in[7];
    const float* bv = (const float*)d_in[8];
    float* out = (float*)d_out;

    const int E = in_sizes[2];
    const int n = in_sizes[0] / HDIM;  // 50000

    // workspace layout: Q | K | V | logits | sum | maxbits
    float* Q      = (float*)d_ws;
    float* K      = Q + (size_t)n * HDIM;
    float* V      = K + (size_t)n * HDIM;
    float* logits = V + (size_t)n * HDIM;
    float* gsum   = logits + E;
    unsigned* gmax = (unsigned*)(gsum + 1);

    hipMemsetAsync(d_out, 0, (size_t)out_size * sizeof(float), stream);
    init_kernel<<<1, 1, 0, stream>>>(gsum, gmax);

    dim3 pgrid((n + 15) / 16, 3);
    qkv_proj_kernel<<<pgrid, 256, 0, stream>>>(x, Wq, bq, Wk, bk, Wv, bv, Q, K, V);

    const int eblocks = (E + 7) / 8;
    edge_logits_kernel<<<eblocks, 256, 0, stream>>>(Q, K, ei, ew, logits, gmax, E);
    exp_sum_kernel<<<(E + 255) / 256, 256, 0, stream>>>(logits, gsum, gmax, E);
    scatter_kernel<<<eblocks, 256, 0, stream>>>(V, logits, gsum, ei, out, E);
}